// CEClassifier_54142357733984
// MI455X (gfx1250) — compile-verified
//
#include <hip/hip_runtime.h>
#include <hip/hip_bf16.h>
#include <math.h>

#define NUM_STEPS 18
#define BATCH 64
#define NIMG 128            // N_CES * B
#define CCH 3
#define HH 64
#define WW 64
#define PLANE (HH * WW)     // 4096
#define IMG (CCH * PLANE)   // 12288
#define NCLS 1000
#define KDIM IMG
#define CE_SIGMA2 0.04f     // 0.2^2
#define SIGMA_DATA2 0.25f   // 0.5^2

typedef __attribute__((ext_vector_type(2))) float v2f;
typedef __attribute__((ext_vector_type(8))) float v8f;

// ---------------------------------------------------------------------------
// Kernel 1: EDM Euler sampler with counterfactual drift. One block per image.
// LDS double-buffers x (2 x 48KB) + mu_ce (48KB) + weights/sigmas.
// ---------------------------------------------------------------------------
__global__ __launch_bounds__(256) void sampler_kernel(
    const float* __restrict__ x,        // [64,3,64,64]
    const float* __restrict__ latents,  // [128,3,64,64]
    const float* __restrict__ noise,    // [18,128,3,64,64] (pre-masked)
    const float* __restrict__ Wn,       // [3,3,3,3] OIHW
    const float* __restrict__ bn,       // [3]
    float* __restrict__ A_out)          // [128,12288] = x_ces
{
    extern __shared__ float lds[];
    float* bufA = lds;                  // 12288
    float* bufB = lds + IMG;            // 12288
    float* mu   = lds + 2 * IMG;        // 12288
    float* wts  = lds + 3 * IMG;        // 81 conv weights + 3 bias
    float* tsh  = wts + 84;             // 19 sigmas

    const int n   = blockIdx.x;
    const int tid = threadIdx.x;

    // Karras schedule in fp64 (reference computes it in float64).
    if (tid < NUM_STEPS + 1) {
        if (tid == NUM_STEPS) {
            tsh[tid] = 0.0f;
        } else {
            const double smax_r = pow(80.0, 1.0 / 7.0);
            const double smin_r = pow(0.002, 1.0 / 7.0);
            const double v = smax_r + (double)tid / (double)(NUM_STEPS - 1) * (smin_r - smax_r);
            tsh[tid] = (float)pow(v, 7.0);
        }
    }
    if (tid < 84) wts[tid] = (tid < 81) ? Wn[tid] : bn[tid - 81];
    __syncthreads();

    const float t0 = tsh[0];
    const float* lat = latents + (size_t)n * IMG;
    const float* xin = x + (size_t)(n % BATCH) * IMG;
    for (int i = tid; i < IMG; i += 256) {
        bufA[i] = lat[i] * t0;
        mu[i]   = 2.0f * xin[i] - 1.0f;
    }
    __syncthreads();

    float* cur = bufA;
    float* nxt = bufB;

    for (int s = 0; s < NUM_STEPS; ++s) {
        const float t   = tsh[s];
        const float tn  = tsh[s + 1];
        const float dt  = t - tn;
        const float s2  = t * t;
        const float den = s2 + SIGMA_DATA2;
        const float rdn = __frsqrt_rn(den);
        const float c_skip = SIGMA_DATA2 / den;
        const float c_out  = t * 0.5f * rdn;
        const float c_in   = rdn;
        const float sn     = sqrtf(2.0f * t * dt);
        const float inv_t  = 1.0f / t;
        const float cf     = t / (CE_SIGMA2 + s2);
        const float* eps = noise + ((size_t)s * NIMG + (size_t)n) * IMG;

        for (int j = 0; j < PLANE / 256; ++j) {
            const int p = tid + j * 256;
            const int h = p >> 6;
            const int w = p & 63;
            float acc0 = 0.0f, acc1 = 0.0f, acc2 = 0.0f;
            #pragma unroll
            for (int kh = 0; kh < 3; ++kh) {
                const int hh = h + kh - 1;
                if ((unsigned)hh >= (unsigned)HH) continue;
                #pragma unroll
                for (int kw = 0; kw < 3; ++kw) {
                    const int ww2 = w + kw - 1;
                    if ((unsigned)ww2 >= (unsigned)WW) continue;
                    const int q = (hh << 6) + ww2;
                    #pragma unroll
                    for (int ci = 0; ci < 3; ++ci) {
                        const float v = cur[ci * PLANE + q];
                        const int wi = ci * 9 + kh * 3 + kw;
                        acc0 = fmaf(wts[0 * 27 + wi], v, acc0);
                        acc1 = fmaf(wts[1 * 27 + wi], v, acc1);
                        acc2 = fmaf(wts[2 * 27 + wi], v, acc2);
                    }
                }
            }
            const float acc[3] = {acc0, acc1, acc2};
            #pragma unroll
            for (int c = 0; c < 3; ++c) {
                const float xc = cur[c * PLANE + p];
                const float f  = c_in * acc[c] + wts[81 + c];        // conv linear: conv(c_in*x)=c_in*conv(x)
                const float dn = c_skip * xc + c_out * f;            // denoised
                const float dx = (dn - xc) * inv_t + cf * (mu[c * PLANE + p] - xc);
                nxt[c * PLANE + p] = xc + 2.0f * dx * dt + sn * eps[c * PLANE + p];
            }
        }
        __syncthreads();
        float* tmp = cur; cur = nxt; nxt = tmp;
    }

    float* dst = A_out + (size_t)n * IMG;
    for (int i = tid; i < IMG; i += 256) dst[i] = (cur[i] + 1.0f) * 0.5f;
}

// ---------------------------------------------------------------------------
// Kernel 2: fp32 WMMA GEMM. logits[128,1000] = A[128,12288] @ Wc[12288,1000] + bc
// One wave32 per 16x16 output tile, K-loop of v_wmma_f32_16x16x4_f32.
// A frag (16x4 f32, 2 VGPR): lane<16 -> row M=lane, K=k..k+1 ; lane>=16 -> K=k+2..k+3
// B frag (4x16 f32, 2 VGPR): lane<16 -> col N=lane, K=k..k+1 ; lane>=16 -> K=k+2..k+3
// C/D (16x16 f32, 8 VGPR): vgpr r -> row (half*8+r), col = lane&15
// ---------------------------------------------------------------------------
__global__ __launch_bounds__(32) void gemm_wmma_kernel(
    const float* __restrict__ A,   // [128, 12288]
    const float* __restrict__ Wc,  // [12288, 1000]
    const float* __restrict__ bc,  // [1000]
    float* __restrict__ L)         // [128, 1000]
{
    const int ntile = blockIdx.x;          // 0..62
    const int mtile = blockIdx.y;          // 0..7
    const int lane  = threadIdx.x;         // 0..31
    const int half  = lane >> 4;           // 0 | 1
    const int l16   = lane & 15;

    const int m     = mtile * 16 + l16;                 // A row for this lane
    const int ncol  = ntile * 16 + l16;                 // output column for this lane
    const int ncolc = ncol < (NCLS - 1) ? ncol : (NCLS - 1); // clamped (no EXEC divergence)

    const float* __restrict__ arow = A + (size_t)m * KDIM;

    v8f c = {0.f, 0.f, 0.f, 0.f, 0.f, 0.f, 0.f, 0.f};

    for (int k = 0; k < KDIM; k += 4) {
        const int ka = k + 2 * half;       // this lane's K base
        v2f a, b;
        a.x = arow[ka];
        a.y = arow[ka + 1];
        b.x = Wc[(size_t)ka * NCLS + ncolc];
        b.y = Wc[(size_t)(ka + 1) * NCLS + ncolc];
        c = __builtin_amdgcn_wmma_f32_16x16x4_f32(
                /*neg_a=*/false, a, /*neg_b=*/false, b,
                /*c_mod=*/(short)0, c, /*reuse_a=*/false, /*reuse_b=*/false);
    }

    if (ncol < NCLS) {
        const float bias = bc[ncol];
        #pragma unroll
        for (int r = 0; r < 8; ++r) {
            const int row = mtile * 16 + half * 8 + r;
            L[(size_t)row * NCLS + ncol] = c[r] + bias;
        }
    }
}

// ---------------------------------------------------------------------------
// Kernel 3: 2-way logsumexp over the CE axis, minus ln(2).
// ---------------------------------------------------------------------------
__global__ __launch_bounds__(256) void lse_kernel(
    const float* __restrict__ L,   // [128, 1000]
    float* __restrict__ out)       // [64, 1000]
{
    const int i = blockIdx.x * blockDim.x + threadIdx.x;
    if (i >= BATCH * NCLS) return;
    const int b = i / NCLS;
    const int j = i - b * NCLS;
    const float l0 = L[(size_t)b * NCLS + j];
    const float l1 = L[(size_t)(b + BATCH) * NCLS + j];
    const float m  = fmaxf(l0, l1);
    out[i] = m + logf(expf(l0 - m) + expf(l1 - m)) - 0.69314718055994531f;
}

// ---------------------------------------------------------------------------
extern "C" void kernel_launch(void* const* d_in, const int* in_sizes, int n_in,
                              void* d_out, int out_size, void* d_ws, size_t ws_size,
                              hipStream_t stream) {
    const float* x       = (const float*)d_in[0];
    const float* latents = (const float*)d_in[1];
    const float* noise   = (const float*)d_in[2];
    const float* Wn      = (const float*)d_in[3];
    const float* bn      = (const float*)d_in[4];
    const float* Wc      = (const float*)d_in[5];
    const float* bc      = (const float*)d_in[6];
    float* out = (float*)d_out;

    float* A = (float*)d_ws;                       // [128, 12288]
    float* L = A + (size_t)NIMG * IMG;             // [128, 1000]

    const size_t shmem = (size_t)(3 * IMG + 128) * sizeof(float);  // ~148 KB of 320 KB/WGP
    hipLaunchKernelGGL(sampler_kernel, dim3(NIMG), dim3(256), shmem, stream,
                       x, latents, noise, Wn, bn, A);

    dim3 gtiles((NCLS + 15) / 16, NIMG / 16);      // 63 x 8 tiles, 1 wave32 each
    hipLaunchKernelGGL(gemm_wmma_kernel, gtiles, dim3(32), 0, stream, A, Wc, bc, L);

    hipLaunchKernelGGL(lse_kernel, dim3((BATCH * NCLS + 255) / 256), dim3(256), 0, stream,
                       L, out);
}